// GraphConvolutionLayer_54485955117401
// MI455X (gfx1250) — compile-verified
//
#include <hip/hip_runtime.h>
#include <hip/hip_bf16.h>

// ---------------------------------------------------------------------------
// GraphConvolutionLayer for MI455X (gfx1250, wave32)
//   h   = X @ W            -> fp32 WMMA (v_wmma_f32_16x16x4_f32), h in d_ws
//   out = relu(scatter_add(w * h[src] -> dst))  -> L2-resident f32 atomics
// ---------------------------------------------------------------------------

typedef __attribute__((ext_vector_type(2))) float v2f;
typedef __attribute__((ext_vector_type(4))) float v4f;
typedef __attribute__((ext_vector_type(8))) float v8f;

#define F_IN  128
#define F_OUT 64
#define MT_PER_BLOCK 4   // M-tiles (16 rows each) per block

// ---------------------------------------------------------------------------
// Kernel 1: h[N,64] = X[N,128] @ W[128,64] via WMMA f32 16x16x4.
// Block = 128 threads = 4 waves. Wave w owns N-tile w (cols w*16..w*16+15).
// Block owns MT_PER_BLOCK consecutive M-tiles. W staged once in LDS (32 KB).
//
// ISA 7.12.2 (32-bit layouts, wave32):
//   A 16x4 : lane L holds row M=(L&15); K = {0,1} for L<16, {2,3} for L>=16
//   B 4x16 : lane L holds col N=(L&15); K = {0,1} for L<16, {2,3} for L>=16
//   C/D    : VGPR j, lane L -> element (M = j + 8*(L>=16), N = L&15)
// ---------------------------------------------------------------------------
__global__ __launch_bounds__(128)
void gcn_gemm_wmma(const float* __restrict__ X,
                   const float* __restrict__ W,
                   float* __restrict__ H,
                   int n_rows, int m_tiles) {
    __shared__ float Wlds[F_IN * F_OUT];   // 32 KB

    const int tid  = threadIdx.x;
    const int wave = tid >> 5;             // 0..3 -> N-tile
    const int lane = tid & 31;

    // Cooperative load of W into LDS as float4 (8192 floats = 2048 float4).
    {
        const v4f* Wv = (const v4f*)W;
        v4f*       Lv = (v4f*)Wlds;
        for (int i = tid; i < (F_IN * F_OUT) / 4; i += 128) {
            Lv[i] = Wv[i];
        }
    }
    __syncthreads();

    const int half = lane >> 4;            // 0: K-pair {0,1}; 1: K-pair {2,3}
    const int ln   = lane & 15;            // M row (for A) / N col (for B/C/D)
    const int n0   = wave * 16;

    for (int mt = 0; mt < MT_PER_BLOCK; ++mt) {
        const int tile = blockIdx.x * MT_PER_BLOCK + mt;
        if (tile >= m_tiles) break;
        const int m0 = tile * 16;

        // Row this lane feeds into the A fragment (clamped for safe loads on
        // a ragged tail; WMMA itself must run with full EXEC).
        int r = m0 + ln;
        const bool r_ok = (r < n_rows);
        if (r >= n_rows) r = n_rows - 1;

        const float* xrow = X + (long long)r * F_IN;

        v8f acc = {};
        #pragma unroll
        for (int k = 0; k < F_IN; k += 4) {
            const int kk = k + 2 * half;
            // A fragment: two consecutive K values of this lane's row.
            v2f a = *(const v2f*)(xrow + kk);
            // B fragment: W[kk][ln], W[kk+1][ln] from LDS.
            v2f b;
            b[0] = Wlds[kk * F_OUT + ln];
            b[1] = Wlds[(kk + 1) * F_OUT + ln];
            // D = A*B + C   (8 args: neg_a, A, neg_b, B, c_mod, C, reuseA, reuseB)
            acc = __builtin_amdgcn_wmma_f32_16x16x4_f32(
                false, a, false, b, (short)0, acc, false, false);
        }

        // Store D: VGPR j, lane -> (row m0 + j + 8*half, col n0 + ln).
        const int rbase = m0 + 8 * half;
        #pragma unroll
        for (int j = 0; j < 8; ++j) {
            const int row = rbase + j;
            if (row < n_rows) {
                H[(long long)row * F_OUT + n0 + ln] = acc[j];
            }
        }
        (void)r_ok;
    }
}

// ---------------------------------------------------------------------------
// Kernel 2: zero the accumulator (d_out is poisoned between runs).
// ---------------------------------------------------------------------------
__global__ void gcn_zero(float* __restrict__ out, int n) {
    int i = blockIdx.x * blockDim.x + threadIdx.x;
    for (; i < n; i += gridDim.x * blockDim.x) out[i] = 0.0f;
}

// ---------------------------------------------------------------------------
// Kernel 3: edge scatter. 32 threads per edge; each lane owns features
// [2f, 2f+1]. Gather of h[src] is a coalesced 256B line; the two f32
// atomic adds land in the 192MB L2 (out is 25.6MB -> fully resident).
// ---------------------------------------------------------------------------
__global__ __launch_bounds__(256)
void gcn_scatter(const float* __restrict__ H,
                 const float* __restrict__ EW,
                 const int* __restrict__ SRC,
                 const int* __restrict__ DST,
                 float* __restrict__ out,
                 int n_edges) {
    const long long tid = (long long)blockIdx.x * blockDim.x + threadIdx.x;
    const int e = (int)(tid >> 5);
    if (e >= n_edges) return;
    const int f = ((int)tid & 31) * 2;

    const int   s = SRC[e];
    const int   d = DST[e];
    const float w = EW[e];

    v2f hv = *(const v2f*)(H + (long long)s * F_OUT + f);
    float* o = out + (long long)d * F_OUT + f;
    atomicAdd(o,     w * hv[0]);
    atomicAdd(o + 1, w * hv[1]);
}

// ---------------------------------------------------------------------------
// Kernel 4: ReLU in place.
// ---------------------------------------------------------------------------
__global__ void gcn_relu(float* __restrict__ out, int n) {
    int i = blockIdx.x * blockDim.x + threadIdx.x;
    for (; i < n; i += gridDim.x * blockDim.x) out[i] = fmaxf(out[i], 0.0f);
}

// ---------------------------------------------------------------------------
// Launch. Inputs (setup_inputs order): x[f32], W[f32], edge_w[f32],
// src[i32], dst[i32]. Output: f32 [N, 64]. d_ws holds h (25.6 MB).
// ---------------------------------------------------------------------------
extern "C" void kernel_launch(void* const* d_in, const int* in_sizes, int n_in,
                              void* d_out, int out_size, void* d_ws, size_t ws_size,
                              hipStream_t stream) {
    const float* x   = (const float*)d_in[0];
    const float* W   = (const float*)d_in[1];
    const float* ew  = (const float*)d_in[2];
    const int*   src = (const int*)d_in[3];
    const int*   dst = (const int*)d_in[4];
    float* out = (float*)d_out;
    float* h   = (float*)d_ws;

    const int n_nodes = out_size / F_OUT;          // 100000
    const int n_edges = in_sizes[2];               // R*E = 3,200,000
    const int m_tiles = (n_nodes + 15) / 16;       // 6250

    // 1) h = X @ W (WMMA fp32)
    {
        dim3 grid((m_tiles + MT_PER_BLOCK - 1) / MT_PER_BLOCK);
        gcn_gemm_wmma<<<grid, 128, 0, stream>>>(x, W, h, n_nodes, m_tiles);
    }
    // 2) out = 0
    {
        int blocks = (out_size + 255) / 256;
        if (blocks > 65535) blocks = 65535;
        gcn_zero<<<blocks, 256, 0, stream>>>(out, out_size);
    }
    // 3) out[dst] += w * h[src]
    {
        const long long total = (long long)n_edges * 32;
        const long long blocks = (total + 255) / 256;
        gcn_scatter<<<dim3((unsigned)blocks), 256, 0, stream>>>(h, ew, src, dst, out, n_edges);
    }
    // 4) out = relu(out)
    {
        int blocks = (out_size + 255) / 256;
        if (blocks > 65535) blocks = 65535;
        gcn_relu<<<blocks, 256, 0, stream>>>(out, out_size);
    }
    (void)n_in; (void)ws_size;
}